// HeteroNewsCompanyGNN_3736621547715
// MI455X (gfx1250) — compile-verified
//
#include <hip/hip_runtime.h>
#include <hip/hip_bf16.h>

typedef __attribute__((ext_vector_type(2))) float v2f;
typedef __attribute__((ext_vector_type(8))) float v8f;

#define H 64          // hidden dim
#define GEMM_TK 64    // K-chunk staged in LDS

// ---------------------------------------------------------------------------
// C[M x 64] = act(A[M x K] @ W[K x 64] + bias)
// One wave = 16 rows x 64 cols via 4x V_WMMA_F32_16X16X4_F32 accumulators.
// Block = 256 threads (8 waves) -> 128 rows/block.
// W chunk staged in LDS pair-interleaved ([k/2][n][k&1]) so each lane's B
// operand is one ds_load_b64; chunk is zero-padded so B reads are unguarded.
// OOB rows clamp to row 0 (harmless garbage accumulate, never stored).
// ---------------------------------------------------------------------------
__global__ __launch_bounds__(256) void gemm_bias_act(
    const float* __restrict__ A, const float* __restrict__ W,
    const float* __restrict__ bias, float* __restrict__ C,
    int M, int K, int doRelu)
{
  __shared__ float sWp[GEMM_TK * H];   // pair-interleaved, 16 KB
  __shared__ float sB[H];
  const int lane  = threadIdx.x & 31;
  const int wave  = threadIdx.x >> 5;
  const int m     = lane & 15;
  const int khalf = (lane >> 4) << 1;        // 0 for lanes 0-15, 2 for 16-31
  const int row0  = blockIdx.x * 128 + wave * 16;
  const int row   = row0 + m;
  const float* arow = A + (size_t)(row < M ? row : 0) * K;

  if (threadIdx.x < H) sB[threadIdx.x] = bias[threadIdx.x];

  v8f acc[4] = {};

  for (int k0 = 0; k0 < K; k0 += GEMM_TK) {
    const int kc = min(GEMM_TK, K - k0);
    __syncthreads();
    // stage + pair-interleave + zero-pad the W chunk
    for (int i = threadIdx.x; i < GEMM_TK * H; i += 256) {
      const int kk = i >> 6;
      const int n  = i & 63;
      const float v = (kk < kc) ? W[(k0 + kk) * H + n] : 0.0f;
      sWp[((kk >> 1) * H + n) * 2 + (kk & 1)] = v;
    }
    __syncthreads();

    if (kc == GEMM_TK) {
      // hot path: no guards anywhere
#pragma unroll 4
      for (int kk = 0; kk < GEMM_TK; kk += 4) {
        const int kb = kk + khalf;
        v2f a;
        a.x = __builtin_nontemporal_load(arow + k0 + kb);
        a.y = __builtin_nontemporal_load(arow + k0 + kb + 1);
#pragma unroll
        for (int t = 0; t < 4; ++t) {
          const v2f b = *(const v2f*)&sWp[((kb >> 1) * H + t * 16 + m) * 2];
          acc[t] = __builtin_amdgcn_wmma_f32_16x16x4_f32(
              false, a, false, b, (short)0, acc[t], false, false);
        }
      }
    } else {
      // tail chunk: guard A only; B already zero-padded
      for (int kk = 0; kk < kc; kk += 4) {
        const int kb = kk + khalf;
        const int kg = k0 + kb;
        v2f a;
        a.x = (kg     < K) ? arow[kg]     : 0.0f;
        a.y = (kg + 1 < K) ? arow[kg + 1] : 0.0f;
#pragma unroll
        for (int t = 0; t < 4; ++t) {
          const v2f b = *(const v2f*)&sWp[((kb >> 1) * H + t * 16 + m) * 2];
          acc[t] = __builtin_amdgcn_wmma_f32_16x16x4_f32(
              false, a, false, b, (short)0, acc[t], false, false);
        }
      }
    }
  }

  if (row0 >= M) return;
  const int rbase = row0 + ((lane >> 4) << 3);   // lanes 16-31 hold rows +8
#pragma unroll
  for (int p = 0; p < 8; ++p) {
    const int r = rbase + p;
    if (r < M) {
#pragma unroll
      for (int t = 0; t < 4; ++t) {
        const int n = t * 16 + m;
        float v = acc[t][p] + sB[n];
        if (doRelu) v = fmaxf(v, 0.0f);
        C[(size_t)r * H + n] = v;
      }
    }
  }
}

// ---------------------------------------------------------------------------
// SAGE combine: out = (S/max(cnt,1)) @ Wl + bl + Xd @ Wr
// Fused K=128 GEMM ([mean | x_dst] @ [Wl;Wr]); weights LDS-resident,
// pair-interleaved. K-loop split into mean-half and x_dst-half (no branches).
// A operands are 8B-aligned -> single global_load_b64 per step.
// NOTE: S and out may alias (per-row reads complete before the row store).
// ---------------------------------------------------------------------------
__global__ __launch_bounds__(256) void sage_combine(
    const float* S, const float* __restrict__ cnt,
    const float* __restrict__ Xd,
    const float* __restrict__ Wl, const float* __restrict__ bl,
    const float* __restrict__ Wr, float* out, int M)
{
  __shared__ float sWp[128 * H];   // pair-interleaved [k/2][n][k&1], 32 KB
  __shared__ float sB[H];
  const int lane  = threadIdx.x & 31;
  const int wave  = threadIdx.x >> 5;
  const int m     = lane & 15;
  const int khalf = (lane >> 4) << 1;
  const int row0  = blockIdx.x * 128 + wave * 16;
  const int row   = row0 + m;
  const int rowc  = row < M ? row : 0;
  const size_t rb = (size_t)rowc * H;
  const float inv = 1.0f / fmaxf(cnt[rowc], 1.0f);

  for (int i = threadIdx.x; i < 64 * H; i += 256) {
    const int kk = i >> 6;
    const int n  = i & 63;
    sWp[((kk >> 1) * H + n) * 2 + (kk & 1)]        = Wl[i];
    sWp[(((kk >> 1) + 32) * H + n) * 2 + (kk & 1)] = Wr[i];
  }
  if (threadIdx.x < H) sB[threadIdx.x] = bl[threadIdx.x];
  __syncthreads();

  v8f acc[4] = {};
  // mean @ Wl
#pragma unroll 4
  for (int k = 0; k < 64; k += 4) {
    const int kb = k + khalf;
    v2f a = *(const v2f*)(S + rb + kb);
    a.x *= inv;
    a.y *= inv;
#pragma unroll
    for (int t = 0; t < 4; ++t) {
      const v2f b = *(const v2f*)&sWp[((kb >> 1) * H + t * 16 + m) * 2];
      acc[t] = __builtin_amdgcn_wmma_f32_16x16x4_f32(
          false, a, false, b, (short)0, acc[t], false, false);
    }
  }
  // x_dst @ Wr
#pragma unroll 4
  for (int k = 64; k < 128; k += 4) {
    const int kb = k + khalf;
    const v2f a = *(const v2f*)(Xd + rb + (kb - 64));
#pragma unroll
    for (int t = 0; t < 4; ++t) {
      const v2f b = *(const v2f*)&sWp[((kb >> 1) * H + t * 16 + m) * 2];
      acc[t] = __builtin_amdgcn_wmma_f32_16x16x4_f32(
          false, a, false, b, (short)0, acc[t], false, false);
    }
  }

  if (row0 >= M) return;
  const int rbase = row0 + ((lane >> 4) << 3);
#pragma unroll
  for (int p = 0; p < 8; ++p) {
    const int r = rbase + p;
    if (r < M) {
#pragma unroll
      for (int t = 0; t < 4; ++t) {
        const int n = t * 16 + m;
        out[(size_t)r * H + n] = acc[t][p] + sB[n];
      }
    }
  }
}

// ---------------------------------------------------------------------------
// Edge scatter: one thread per (edge, channel). Feature tables are L2-resident
// (25.6 MB << 192 MB L2) so atomics/gathers stay on-chip.
// ---------------------------------------------------------------------------
__global__ void scatter_add64(const float* __restrict__ X,
                              const int* __restrict__ src,
                              const int* __restrict__ dst,
                              float* __restrict__ sums,
                              float* cnt, long long nE)
{
  const long long tid = (long long)blockIdx.x * blockDim.x + threadIdx.x;
  const long long e = tid >> 6;
  if (e >= nE) return;
  const int h = (int)(tid & 63);
  const int s = src[e];
  const int d = dst[e];
  atomicAdd(&sums[(size_t)d * H + h], X[(size_t)s * H + h]);
  if (h == 0 && cnt) atomicAdd(&cnt[d], 1.0f);
}

// ---------------------------------------------------------------------------
// out = LayerNorm(relu(scale * (A (+ B)))), per 64-channel row; wave per row.
// ---------------------------------------------------------------------------
__global__ __launch_bounds__(256) void avg_relu_ln(
    const float* A, const float* B, float scale,
    const float* __restrict__ g, const float* __restrict__ bb,
    float* out, int M)
{
  const int lane = threadIdx.x & 31;
  const int row  = blockIdx.x * 8 + (threadIdx.x >> 5);
  if (row >= M) return;
  const size_t rb = (size_t)row * H;
  float v0 = A[rb + lane], v1 = A[rb + lane + 32];
  if (B) { v0 += B[rb + lane]; v1 += B[rb + lane + 32]; }
  v0 = fmaxf(v0 * scale, 0.0f);
  v1 = fmaxf(v1 * scale, 0.0f);
  float s = v0 + v1;
#pragma unroll
  for (int off = 16; off > 0; off >>= 1) s += __shfl_xor(s, off, 32);
  const float mean = s * (1.0f / 64.0f);
  const float d0 = v0 - mean, d1 = v1 - mean;
  float q = d0 * d0 + d1 * d1;
#pragma unroll
  for (int off = 16; off > 0; off >>= 1) q += __shfl_xor(q, off, 32);
  const float rstd = rsqrtf(q * (1.0f / 64.0f) + 1e-5f);
  out[rb + lane]      = d0 * rstd * g[lane]      + bb[lane];
  out[rb + lane + 32] = d1 * rstd * g[lane + 32] + bb[lane + 32];
}

// ---------------------------------------------------------------------------
// Classifier head: relu(x@W1+b1)@W2+b2, one thread per row (tiny: 2000 rows).
// ---------------------------------------------------------------------------
__global__ __launch_bounds__(256) void cls_head(
    const float* __restrict__ X, const float* __restrict__ W1,
    const float* __restrict__ b1, const float* __restrict__ W2,
    const float* __restrict__ b2, float* __restrict__ out, int M)
{
  __shared__ float sW1[64 * 32];
  __shared__ float sb1[32];
  __shared__ float sW2[32];
  __shared__ float sb2;
  for (int i = threadIdx.x; i < 64 * 32; i += blockDim.x) sW1[i] = W1[i];
  if (threadIdx.x < 32) { sb1[threadIdx.x] = b1[threadIdx.x]; sW2[threadIdx.x] = W2[threadIdx.x]; }
  if (threadIdx.x == 0) sb2 = b2[0];
  __syncthreads();
  const int row = blockIdx.x * blockDim.x + threadIdx.x;
  if (row >= M) return;
  const float* x = X + (size_t)row * H;
  float h[64];
#pragma unroll
  for (int k = 0; k < 64; ++k) h[k] = x[k];
  float acc = sb2;
  for (int j = 0; j < 32; ++j) {
    float t = sb1[j];
#pragma unroll
    for (int k = 0; k < 64; ++k) t += h[k] * sW1[k * 32 + j];
    acc += fmaxf(t, 0.0f) * sW2[j];
  }
  out[row] = acc;
}

// ---------------------------------------------------------------------------
extern "C" void kernel_launch(void* const* d_in, const int* in_sizes, int n_in,
                              void* d_out, int out_size, void* d_ws, size_t ws_size,
                              hipStream_t stream)
{
  const float* x_news = (const float*)d_in[0];
  const float* x_comp = (const float*)d_in[1];
  const int* sim_src = (const int*)d_in[2];
  const int* sim_dst = (const int*)d_in[3];
  const int* men_src = (const int*)d_in[4];
  const int* men_dst = (const int*)d_in[5];
  const int* rel_src = (const int*)d_in[6];
  const int* rel_dst = (const int*)d_in[7];
  // params flattened as a JAX pytree (dict keys sorted at every level):
  // cls{W1,W2,b1,b2}, comp_proj{W,b}, conv1{men,rel,sim}{Wl,Wr,bl},
  // conv2{men,rel,sim}{Wl,Wr,bl}, ln1_comp{b,g}, ln1_news{b,g}, ln2_comp{b,g},
  // news_proj{W,b}
  const float* cls_W1 = (const float*)d_in[8];
  const float* cls_W2 = (const float*)d_in[9];
  const float* cls_b1 = (const float*)d_in[10];
  const float* cls_b2 = (const float*)d_in[11];
  const float* cp_W   = (const float*)d_in[12];
  const float* cp_b   = (const float*)d_in[13];
  const float* c1_men_Wl = (const float*)d_in[14];
  const float* c1_men_Wr = (const float*)d_in[15];
  const float* c1_men_bl = (const float*)d_in[16];
  const float* c1_rel_Wl = (const float*)d_in[17];
  const float* c1_rel_Wr = (const float*)d_in[18];
  const float* c1_rel_bl = (const float*)d_in[19];
  const float* c1_sim_Wl = (const float*)d_in[20];
  const float* c1_sim_Wr = (const float*)d_in[21];
  const float* c1_sim_bl = (const float*)d_in[22];
  const float* c2_men_Wl = (const float*)d_in[23];
  const float* c2_men_Wr = (const float*)d_in[24];
  const float* c2_men_bl = (const float*)d_in[25];
  const float* c2_rel_Wl = (const float*)d_in[26];
  const float* c2_rel_Wr = (const float*)d_in[27];
  const float* c2_rel_bl = (const float*)d_in[28];
  // d_in[29..31] = conv2.sim (unused: conv2 news output is elided upstream)
  const float* ln1c_b = (const float*)d_in[32];
  const float* ln1c_g = (const float*)d_in[33];
  const float* ln1n_b = (const float*)d_in[34];
  const float* ln1n_g = (const float*)d_in[35];
  const float* ln2c_b = (const float*)d_in[36];
  const float* ln2c_g = (const float*)d_in[37];
  const float* np_W = (const float*)d_in[38];
  const float* np_b = (const float*)d_in[39];

  const int K_NEWS = 385, K_COMP = 24;
  const int N_NEWS = in_sizes[0] / K_NEWS;   // 100000
  const int N_COMP = in_sizes[1] / K_COMP;   // 2000
  const long long E_SIM = in_sizes[2];
  const long long E_MEN = in_sizes[4];
  const long long E_REL = in_sizes[6];

  float* w = (float*)d_ws;
  size_t off = 0;
  auto alloc = [&](size_t n) { size_t o = off; off += (n + 63) & ~(size_t)63; return o; };

  // --- zeroed region (one contiguous memset) ---
  const size_t z0   = off;
  const size_t sumN = alloc((size_t)N_NEWS * H);   // conv1 sim sums; reused as n1/hn1
  const size_t cntN = alloc((size_t)N_NEWS);
  const size_t sumM = alloc((size_t)N_COMP * H);
  const size_t cntM = alloc((size_t)N_COMP);       // mention degree (reused in conv2)
  const size_t sumR = alloc((size_t)N_COMP * H);
  const size_t cntR = alloc((size_t)N_COMP);       // related degree (reused in conv2)
  const size_t sumM2 = alloc((size_t)N_COMP * H);
  const size_t sumR2 = alloc((size_t)N_COMP * H);
  const size_t z1   = off;
  // --- fully-overwritten region ---
  const size_t hn   = alloc((size_t)N_NEWS * H);
  const size_t hc   = alloc((size_t)N_COMP * H);
  const size_t coA  = alloc((size_t)N_COMP * H);
  const size_t coB  = alloc((size_t)N_COMP * H);
  const size_t hc1  = alloc((size_t)N_COMP * H);
  const size_t co2A = alloc((size_t)N_COMP * H);
  const size_t co2B = alloc((size_t)N_COMP * H);
  const size_t hc2  = alloc((size_t)N_COMP * H);
  (void)ws_size; (void)n_in; (void)out_size;

  hipMemsetAsync(w + z0, 0, (z1 - z0) * sizeof(float), stream);

  const dim3 blk(256);
  auto gemmGrid    = [](int M) { return dim3((unsigned)((M + 127) / 128)); };
  auto scatterGrid = [](long long e) { return dim3((unsigned)((e * 64 + 255) / 256)); };
  auto rowGrid     = [](int M) { return dim3((unsigned)((M + 7) / 8)); };

  // Projections: hn = relu(x_news@Wn+bn), hc = relu(x_comp@Wc+bc)
  gemm_bias_act<<<gemmGrid(N_NEWS), blk, 0, stream>>>(x_news, np_W, np_b, w + hn, N_NEWS, K_NEWS, 1);
  gemm_bias_act<<<gemmGrid(N_COMP), blk, 0, stream>>>(x_comp, cp_W, cp_b, w + hc, N_COMP, K_COMP, 1);

  // conv1 scatter-sums + degrees
  scatter_add64<<<scatterGrid(E_SIM), blk, 0, stream>>>(w + hn, sim_src, sim_dst, w + sumN, w + cntN, E_SIM);
  scatter_add64<<<scatterGrid(E_MEN), blk, 0, stream>>>(w + hn, men_src, men_dst, w + sumM, w + cntM, E_MEN);
  scatter_add64<<<scatterGrid(E_REL), blk, 0, stream>>>(w + hc, rel_src, rel_dst, w + sumR, w + cntR, E_REL);

  // conv1 combines (news in place over its own sum buffer)
  sage_combine<<<gemmGrid(N_NEWS), blk, 0, stream>>>(w + sumN, w + cntN, w + hn, c1_sim_Wl, c1_sim_bl, c1_sim_Wr, w + sumN, N_NEWS);
  sage_combine<<<gemmGrid(N_COMP), blk, 0, stream>>>(w + sumM, w + cntM, w + hc, c1_men_Wl, c1_men_bl, c1_men_Wr, w + coA, N_COMP);
  sage_combine<<<gemmGrid(N_COMP), blk, 0, stream>>>(w + sumR, w + cntR, w + hc, c1_rel_Wl, c1_rel_bl, c1_rel_Wr, w + coB, N_COMP);

  // hn1 = LN(relu(n1)) in place; hc1 = LN(relu(0.5*(men+rel)))
  avg_relu_ln<<<rowGrid(N_NEWS), blk, 0, stream>>>(w + sumN, nullptr, 1.0f, ln1n_g, ln1n_b, w + sumN, N_NEWS);
  avg_relu_ln<<<rowGrid(N_COMP), blk, 0, stream>>>(w + coA, w + coB, 0.5f, ln1c_g, ln1c_b, w + hc1, N_COMP);

  // conv2 scatter-sums (degrees unchanged -> reuse cntM/cntR)
  scatter_add64<<<scatterGrid(E_MEN), blk, 0, stream>>>(w + sumN, men_src, men_dst, w + sumM2, nullptr, E_MEN);
  scatter_add64<<<scatterGrid(E_REL), blk, 0, stream>>>(w + hc1, rel_src, rel_dst, w + sumR2, nullptr, E_REL);

  sage_combine<<<gemmGrid(N_COMP), blk, 0, stream>>>(w + sumM2, w + cntM, w + hc1, c2_men_Wl, c2_men_bl, c2_men_Wr, w + co2A, N_COMP);
  sage_combine<<<gemmGrid(N_COMP), blk, 0, stream>>>(w + sumR2, w + cntR, w + hc1, c2_rel_Wl, c2_rel_bl, c2_rel_Wr, w + co2B, N_COMP);

  avg_relu_ln<<<rowGrid(N_COMP), blk, 0, stream>>>(w + co2A, w + co2B, 0.5f, ln2c_g, ln2c_b, w + hc2, N_COMP);

  cls_head<<<dim3((unsigned)((N_COMP + 255) / 256)), blk, 0, stream>>>(
      w + hc2, cls_W1, cls_b1, cls_W2, cls_b2, (float*)d_out, N_COMP);
}